// NMS_89094801588313
// MI455X (gfx1250) — compile-verified
//
#include <hip/hip_runtime.h>
#include <hip/hip_bf16.h>
#include <stdint.h>

// ---------------- problem constants ----------------
#define NBATCH   16
#define NPRED    25200
#define NCH      85
#define NCLS     80
#define TOPK     2048
#define MAXDET   1000
#define NCAND    4096
#define CONF_T   0.25f
#define IOU_T    0.45f
#define MAX_WH   4096.0f
// iou > t  <=>  inter > k*(area_i + area_j + eps),  k = t/(1+t)
#define K_IOU    (IOU_T / (1.0f + IOU_T))
#define K_EPS    (K_IOU * 1e-7f)

typedef __attribute__((ext_vector_type(2))) float v2f;
typedef __attribute__((ext_vector_type(8))) float v8f;

__device__ __forceinline__ unsigned ballot32(bool p) {
#if __has_builtin(__builtin_amdgcn_ballot_w32)
  return __builtin_amdgcn_ballot_w32(p);
#else
  return (unsigned)__ballot(p);
#endif
}

__device__ __forceinline__ void wait_asynccnt0() {
#if __has_builtin(__builtin_amdgcn_s_wait_asynccnt)
  __builtin_amdgcn_s_wait_asynccnt(0);
#else
  asm volatile("s_wait_asynccnt 0" ::: "memory");
#endif
}

// ---------------- stage 1: per-prediction score + argmax ----------------
__global__ __launch_bounds__(256)
void k_score(const float* __restrict__ pred,
             float* __restrict__ score, int* __restrict__ cls) {
  int gid = blockIdx.x * blockDim.x + threadIdx.x;
  if (gid >= NBATCH * NPRED) return;
  const float* row = pred + (size_t)gid * NCH;
  // pull tail cachelines of the 340B row early (global_prefetch_b8)
  __builtin_prefetch(row + 32, 0, 1);
  __builtin_prefetch(row + 64, 0, 1);
  float obj = row[4];
  float best = row[5];
  int bi = 0;
#pragma unroll 4
  for (int c = 1; c < NCLS; ++c) {
    float v = row[5 + c];
    if (v > best) { best = v; bi = c; }   // first-max wins (strict >) like argmax
  }
  float sc = obj * best;
  bool valid = (obj > CONF_T) && (sc > CONF_T);
  score[gid] = valid ? sc : 0.0f;
  cls[gid] = bi;
}

// ---------------- stage 2: per-batch top-2048 (hist cutoff + bitonic) ----------------
__global__ __launch_bounds__(1024)
void k_topk(const float* __restrict__ score,
            float* __restrict__ tk_score, unsigned* __restrict__ tk_idx) {
  const int b = blockIdx.x;
  const int tid = threadIdx.x;
  const float* s = score + (size_t)b * NPRED;

  __shared__ unsigned hist[1024];
  __shared__ unsigned long long cand[NCAND];
  __shared__ int s_cut;
  __shared__ unsigned s_cnt;

  hist[tid] = 0;
  if (tid == 0) s_cnt = 0;
  for (int e = tid; e < NCAND; e += 1024) cand[e] = 0ULL;
  __syncthreads();

  // histogram of positive scores, 1024 bins over [0,1]
  for (int p = tid; p < NPRED; p += 1024) {
    float v = s[p];
    if (v > 0.0f) {
      int bin = (int)(v * 1024.0f);
      bin = bin > 1023 ? 1023 : bin;
      atomicAdd(&hist[bin], 1u);
    }
  }
  __syncthreads();

  if (tid == 0) {
    unsigned cum = 0; int cut = 0;
    for (int bin = 1023; bin >= 0; --bin) {
      cum += hist[bin];
      if (cum >= TOPK) { cut = bin; break; }
    }
    s_cut = cut;
  }
  __syncthreads();
  const int cut = s_cut;

  // compact candidates above cutoff; key = score_bits<<32 | ~idx (stable ties)
  for (int p = tid; p < NPRED; p += 1024) {
    float v = s[p];
    if (v > 0.0f) {
      int bin = (int)(v * 1024.0f);
      bin = bin > 1023 ? 1023 : bin;
      if (bin >= cut) {
        unsigned pos = atomicAdd(&s_cnt, 1u);
        if (pos < NCAND) {
          unsigned sb = __float_as_uint(v);
          cand[pos] = ((unsigned long long)sb << 32) | (unsigned)(~(unsigned)p);
        }
      }
    }
  }
  __syncthreads();

  // bitonic sort, descending
  for (unsigned k = 2; k <= NCAND; k <<= 1) {
    for (unsigned j = k >> 1; j > 0; j >>= 1) {
      for (unsigned e = tid; e < NCAND; e += 1024) {
        unsigned partner = e ^ j;
        if (partner > e) {
          unsigned long long x = cand[e], y = cand[partner];
          bool blockAsc = ((e & k) == 0);        // invert -> descending final
          bool doSwap = blockAsc ? (x < y) : (x > y);
          if (doSwap) { cand[e] = y; cand[partner] = x; }
        }
      }
      __syncthreads();
    }
  }

  for (int o = tid; o < TOPK; o += 1024) {
    unsigned long long key = cand[o];
    unsigned sb = (unsigned)(key >> 32);
    float v = __uint_as_float(sb);
    unsigned idx = ~(unsigned)(key & 0xFFFFFFFFu);
    tk_score[(size_t)b * TOPK + o] = v;
    tk_idx[(size_t)b * TOPK + o] = (v > 0.0f) ? idx : 0u;
  }
}

// ---------------- stage 3: gather boxes / shifted boxes / areas ----------------
__global__ __launch_bounds__(256)
void k_gather(const float* __restrict__ pred,
              const int* __restrict__ cls,
              const unsigned* __restrict__ tk_idx,
              float* __restrict__ boxk, float* __restrict__ shifted,
              float* __restrict__ area, float* __restrict__ clsk) {
  int gid = blockIdx.x * blockDim.x + threadIdx.x;
  if (gid >= NBATCH * TOPK) return;
  int b = gid >> 11;
  unsigned idx = tk_idx[gid];
  const float* row = pred + ((size_t)b * NPRED + idx) * NCH;
  float cx = row[0], cy = row[1], w = row[2], h = row[3];
  float x1 = cx - w * 0.5f, y1 = cy - h * 0.5f;
  float x2 = cx + w * 0.5f, y2 = cy + h * 0.5f;
  int c = cls[(size_t)b * NPRED + idx];
  float shift = (float)c * MAX_WH;
  boxk[(size_t)gid * 4 + 0] = x1;
  boxk[(size_t)gid * 4 + 1] = y1;
  boxk[(size_t)gid * 4 + 2] = x2;
  boxk[(size_t)gid * 4 + 3] = y2;
  shifted[(size_t)gid * 4 + 0] = x1 + shift;
  shifted[(size_t)gid * 4 + 1] = y1 + shift;
  shifted[(size_t)gid * 4 + 2] = x2 + shift;
  shifted[(size_t)gid * 4 + 3] = y2 + shift;
  area[gid] = (x2 - x1) * (y2 - y1);
  clsk[gid] = (float)c;
}

// ---------------- stage 4: IoU tiles -> suppression bitmask ----------------
// One block = (batch, 16-row i-tile). 8 waves sweep the 128 j-tiles.
// WMMA f32 16x16x4 computes k*(area_i + area_j + eps) as a rank-2 matmul:
//   A(16x4) = [k*area_i + k*eps, k, 0, 0],  B(4x16) = [1; area_j; 0; 0]
// so the per-element IoU test collapses to inter > D[i][j].
__global__ __launch_bounds__(256)
void k_iou(const float* __restrict__ shifted, const float* __restrict__ area,
           unsigned short* __restrict__ mask16) {
  const int b  = blockIdx.x >> 7;
  const int it = blockIdx.x & 127;
  const int i0 = it * 16;
  const int tid = threadIdx.x;
  const int wave = tid >> 5;
  const int lane = tid & 31;

  __shared__ __align__(16) float rbox[16][4];
  __shared__ __align__(16) float rarea[16];

  // stage the 16 i-rows into LDS with async global->LDS copies (ASYNCcnt path)
  if (tid < 16) {
    const float* gb = shifted + ((size_t)b * TOPK + i0 + tid) * 4;
    const float* ga = area + ((size_t)b * TOPK + i0 + tid);
    unsigned lds_b = (unsigned)(uintptr_t)&rbox[tid][0];   // flat LDS addr truncates to LDS offset
    unsigned lds_a = (unsigned)(uintptr_t)&rarea[tid];
    asm volatile("global_load_async_to_lds_b128 %0, %1, off"
                 :: "v"(lds_b), "v"(gb) : "memory");
    asm volatile("global_load_async_to_lds_b32 %0, %1, off"
                 :: "v"(lds_a), "v"(ga) : "memory");
  }
  wait_asynccnt0();
  __syncthreads();

  v2f a;
  a.x = (lane < 16) ? fmaf(K_IOU, rarea[lane & 15], K_EPS) : 0.0f; // K=0: k*area_i + k*eps
  a.y = (lane < 16) ? K_IOU : 0.0f;                                // K=1: k

  for (int jt = wave; jt < 128; jt += 8) {
    const int j0 = jt * 16;
    const int jl = j0 + (lane & 15);
    const float4 jb = *(const float4*)(shifted + ((size_t)b * TOPK + jl) * 4);
    const float aj = area[(size_t)b * TOPK + jl];

    v2f bm;
    bm.x = (lane < 16) ? 1.0f : 0.0f;  // K=0 row: ones
    bm.y = (lane < 16) ? aj   : 0.0f;  // K=1 row: area_j
    v8f c = {};
    // D[i][j] = k*(area_i + area_j + eps)
    v8f d = __builtin_amdgcn_wmma_f32_16x16x4_f32(false, a, false, bm,
                                                  (short)0, c, false, false);
    unsigned val = 0;
#pragma unroll
    for (int r = 0; r < 8; ++r) {
      int ir = (lane < 16) ? r : (r + 8);
      float ix1 = rbox[ir][0], iy1 = rbox[ir][1];
      float ix2 = rbox[ir][2], iy2 = rbox[ir][3];
      float ltx = fmaxf(ix1, jb.x), lty = fmaxf(iy1, jb.y);
      float rbx = fminf(ix2, jb.z), rby = fminf(iy2, jb.w);
      float iw = fmaxf(rbx - ltx, 0.0f);
      float ih = fmaxf(rby - lty, 0.0f);
      float inter = iw * ih;
      bool p = inter > d[r];          // iou > IOU_T (threshold folded into D)
      unsigned m = ballot32(p);       // low16: row i0+r, high16: row i0+8+r
      val = (lane == r) ? (m & 0xFFFFu)
          : (lane == (r + 8)) ? (m >> 16) : val;
    }
    if (lane < 16) {
      // lanes 0..7 -> rows i0+lane (low halves); lanes 8..15 -> rows i0+lane (high halves)
      mask16[((size_t)b * TOPK + i0 + lane) * 128 + jt] = (unsigned short)val;
    }
  }
}

// ---------------- stage 5: greedy suppression + compaction ----------------
__global__ __launch_bounds__(64)
void k_suppress(const float* __restrict__ tk_score,
                const unsigned* __restrict__ mask32,
                const float* __restrict__ boxk, const float* __restrict__ clsk,
                float* __restrict__ out) {
  const int b = blockIdx.x;
  const int t = threadIdx.x;   // 64 threads, thread t owns keep word t (32 i's)

  __shared__ unsigned keep[64];
  __shared__ unsigned cnt[64];
  __shared__ unsigned off[65];

  // init keep = val_k  (score > 0)
  {
    unsigned w = 0;
    for (int p = 0; p < 32; ++p) {
      int i = t * 32 + p;
      if (tk_score[(size_t)b * TOPK + i] > 0.0f) w |= (1u << p);
    }
    keep[t] = w;
  }
  __syncthreads();

  const unsigned* mrows = mask32 + (size_t)b * TOPK * 64;
  for (int i = 0; i < TOPK; ++i) {
    unsigned kw = keep[i >> 5];
    bool on = (kw >> (i & 31)) & 1u;
    __syncthreads();
    if (on) {
      unsigned m = mrows[(size_t)i * 64 + t];
      int iw = i >> 5;
      if (t < iw) m = 0;
      else if (t == iw) m &= ~(unsigned)((1ULL << ((i & 31) + 1)) - 1ULL); // j > i only
      keep[t] &= ~m;
    }
    __syncthreads();
  }

  // compaction
  cnt[t] = __popc(keep[t]);
  __syncthreads();
  if (t == 0) {
    unsigned run = 0;
    for (int w = 0; w < 64; ++w) { off[w] = run; run += cnt[w]; }
    off[64] = run;
  }
  __syncthreads();

  const unsigned total = off[64];
  {
    unsigned base = off[t];
    unsigned w = keep[t];
    unsigned local = 0;
    for (int p = 0; p < 32; ++p) {
      if ((w >> p) & 1u) {
        unsigned r = base + local;
        ++local;
        if (r < MAXDET) {
          size_t gi = (size_t)b * TOPK + (t * 32 + p);
          float* orow = out + ((size_t)b * MAXDET + r) * 6;
          orow[0] = boxk[gi * 4 + 0];
          orow[1] = boxk[gi * 4 + 1];
          orow[2] = boxk[gi * 4 + 2];
          orow[3] = boxk[gi * 4 + 3];
          orow[4] = tk_score[gi];
          orow[5] = clsk[gi];
        }
      }
    }
  }
  // padding rows: [0,0,0,0,0,-1]
  for (unsigned r = total + t; r < MAXDET; r += 64) {
    float* orow = out + ((size_t)b * MAXDET + r) * 6;
    orow[0] = 0.0f; orow[1] = 0.0f; orow[2] = 0.0f; orow[3] = 0.0f;
    orow[4] = 0.0f; orow[5] = -1.0f;
  }
}

// ---------------- launch ----------------
extern "C" void kernel_launch(void* const* d_in, const int* in_sizes, int n_in,
                              void* d_out, int out_size, void* d_ws, size_t ws_size,
                              hipStream_t stream) {
  const float* pred = (const float*)d_in[0];
  float* out = (float*)d_out;
  char* ws = (char*)d_ws;

  // workspace layout (256B aligned)
  constexpr size_t off_score = 0;                                        // 16*25200 f32
  constexpr size_t off_cls   = off_score + (size_t)NBATCH * NPRED * 4;   // 16*25200 i32
  constexpr size_t off_tks   = off_cls   + (size_t)NBATCH * NPRED * 4;   // 16*2048 f32
  constexpr size_t off_tki   = off_tks   + (size_t)NBATCH * TOPK * 4;    // 16*2048 u32
  constexpr size_t off_boxk  = off_tki   + (size_t)NBATCH * TOPK * 4;    // 16*2048*4 f32
  constexpr size_t off_shift = off_boxk  + (size_t)NBATCH * TOPK * 16;   // 16*2048*4 f32
  constexpr size_t off_area  = off_shift + (size_t)NBATCH * TOPK * 16;   // 16*2048 f32
  constexpr size_t off_clsk  = off_area  + (size_t)NBATCH * TOPK * 4;    // 16*2048 f32
  constexpr size_t off_mask  = off_clsk  + (size_t)NBATCH * TOPK * 4;    // 16*2048*128 u16

  float*    score   = (float*)(ws + off_score);
  int*      cls     = (int*)(ws + off_cls);
  float*    tks     = (float*)(ws + off_tks);
  unsigned* tki     = (unsigned*)(ws + off_tki);
  float*    boxk    = (float*)(ws + off_boxk);
  float*    shifted = (float*)(ws + off_shift);
  float*    area    = (float*)(ws + off_area);
  float*    clsk    = (float*)(ws + off_clsk);
  unsigned short* mask16 = (unsigned short*)(ws + off_mask);
  unsigned*       mask32 = (unsigned*)(ws + off_mask);

  k_score<<<(NBATCH * NPRED + 255) / 256, 256, 0, stream>>>(pred, score, cls);
  k_topk<<<NBATCH, 1024, 0, stream>>>(score, tks, tki);
  k_gather<<<(NBATCH * TOPK + 255) / 256, 256, 0, stream>>>(pred, cls, tki,
                                                            boxk, shifted, area, clsk);
  k_iou<<<NBATCH * 128, 256, 0, stream>>>(shifted, area, mask16);
  k_suppress<<<NBATCH, 64, 0, stream>>>(tks, mask32, boxk, clsk, out);
}